// GraphEncoder_17798344475242
// MI455X (gfx1250) — compile-verified
//
#include <hip/hip_runtime.h>
#include <math.h>

#define N_ATOMS  30
#define DIM_NODE 27
#define DIM_EDGE 12
#define DIM_VSA  1024
#define MAX_EDGES 70
#define N2 (N_ATOMS * N_ATOMS)          // 900
#define KDIM 40                          // 27 + 12 padded to 40 (col 39 == 0)
#define MPAD 80                          // 70 rows padded to 80 (5 M-tiles of 16)

typedef __attribute__((ext_vector_type(2))) float v2f;
typedef __attribute__((ext_vector_type(8))) float v8f;

__global__ __launch_bounds__(256)
__attribute__((amdgpu_waves_per_eu(4)))   // cap at 256 VGPRs: no s_set_vgpr_msb, 4 waves/SIMD
void graph_encode_kernel(const float* __restrict__ atoms,
                         const float* __restrict__ pairs,
                         const unsigned char* __restrict__ mask,
                         const float* __restrict__ atomP,
                         const float* __restrict__ bondP,
                         const float* __restrict__ freq,
                         float* __restrict__ out)
{
    __shared__ float         s_atoms[N_ATOMS * DIM_NODE];   // 810 f32
    __shared__ float         s_F[MPAD * KDIM];              // 80x40 f32
    __shared__ unsigned char s_mask[N2];
    __shared__ int           s_edges[MAX_EDGES];
    __shared__ int           s_cnt;

    const int b   = blockIdx.x;
    const int tid = threadIdx.x;

    const float*         atoms_g = atoms + (size_t)b * N_ATOMS * DIM_NODE;
    const float*         pairs_g = pairs + (size_t)b * N2 * DIM_EDGE;
    const unsigned char* mask_g  = mask  + (size_t)b * N2;

    // ---- stage atoms + mask into LDS ----
    for (int i = tid; i < N_ATOMS * DIM_NODE; i += 256) s_atoms[i] = atoms_g[i];
    for (int i = tid; i < N2; i += 256)                 s_mask[i]  = mask_g[i];
    __syncthreads();

    // ---- wave 0: compact upper-tri active edges (row-major stable order) ----
    if (tid < 32) {
        int cnt = 0;
        for (int c = 0; c < (N2 + 31) / 32; ++c) {
            int pos = c * 32 + tid;
            bool pred = false;
            if (pos < N2) {
                int i = pos / N_ATOMS;
                int j = pos - i * N_ATOMS;
                pred = (j >= i) && (s_mask[pos] != 0);
            }
            unsigned bal = (unsigned)__ballot(pred);     // wave32: low 32 bits
            int ofs  = __popc(bal & ((1u << tid) - 1u));
            int slot = cnt + ofs;
            if (pred && slot < MAX_EDGES) s_edges[slot] = pos;
            cnt += __popc(bal);
        }
        if (tid == 0) s_cnt = cnt < MAX_EDGES ? cnt : MAX_EDGES;
    }
    __syncthreads();

    const int nedge = s_cnt;

    // ---- build fused feature matrix F[80][40]:
    //      cols 0..26 = atoms[src]+atoms[dst], 27..38 = pairs[pos], 39 = 0 ----
    for (int e = tid; e < MPAD * KDIM; e += 256) {
        int m = e / KDIM;
        int k = e - m * KDIM;
        float v = 0.f;
        if (m < nedge) {
            int pos = s_edges[m];
            if (k < DIM_NODE) {
                int src = pos / N_ATOMS;
                int dst = pos - src * N_ATOMS;
                v = s_atoms[src * DIM_NODE + k] + s_atoms[dst * DIM_NODE + k];
            } else if (k < DIM_NODE + DIM_EDGE) {
                v = pairs_g[(size_t)pos * DIM_EDGE + (k - DIM_NODE)];
            }
        }
        s_F[e] = v;
    }
    __syncthreads();

    // ---- WMMA GEMM + trig epilogue. Each wave owns a 128-column strip. ----
    const int wid  = tid >> 5;
    const int lane = tid & 31;
    const int cl   = lane & 15;      // N (and M) index within tile
    const int hi   = lane >> 4;      // half-wave select

    float* out_g = out + (size_t)b * MAX_EDGES * DIM_VSA;
    const float INV_PI = 0.318309886183791f;
    const float PI_F   = 3.14159265358979f;

    for (int nt = 0; nt < 8; ++nt) {
        const int n0  = (wid * 8 + nt) * 16;
        const int col = n0 + cl;
        const float fv = freq[col];

        // B fragments for all 10 K-steps (4x16 f32 layout: lane=N, hi half = K+2)
        v2f Bf[10];
#pragma unroll
        for (int kt = 0; kt < 10; ++kt) {
            int r0 = kt * 4 + hi * 2;
            int r1 = r0 + 1;
            if (r1 > 38) r1 = 38;    // logical K=39 is dead: F[:,39] == 0
            const float* p0 = (r0 < DIM_NODE) ? (atomP + (size_t)r0 * DIM_VSA)
                                              : (bondP + (size_t)(r0 - DIM_NODE) * DIM_VSA);
            const float* p1 = (r1 < DIM_NODE) ? (atomP + (size_t)r1 * DIM_VSA)
                                              : (bondP + (size_t)(r1 - DIM_NODE) * DIM_VSA);
            Bf[kt].x = p0[col];
            Bf[kt].y = p1[col];
        }

#pragma unroll
        for (int mt = 0; mt < 5; ++mt) {
            const int m0 = mt * 16;
            v8f acc = {0.f, 0.f, 0.f, 0.f, 0.f, 0.f, 0.f, 0.f};
            // A fragment base: lane = M row, hi half = K+2 (aligned b64 reads)
            const float* fr = &s_F[(m0 + cl) * KDIM + hi * 2];
#pragma unroll
            for (int kt = 0; kt < 10; ++kt) {
                v2f Af;
                Af.x = fr[kt * 4 + 0];
                Af.y = fr[kt * 4 + 1];
                acc = __builtin_amdgcn_wmma_f32_16x16x4_f32(
                          false, Af, false, Bf[kt], (short)0, acc, false, false);
            }
            // epilogue: D layout -> VGPR r, lanes 0-15: M=m0+r, lanes 16-31: M=m0+8+r
#pragma unroll
            for (int r = 0; r < 8; ++r) {
                int m = m0 + r + 8 * hi;
                if (m >= MAX_EDGES) continue;
                float g = 0.f;
                if (m < nedge) {
                    // bound phase (pi units) = acc[r]; trig is 2-periodic so no remap needed
                    float xb = PI_F * acc[r];
                    float xs = PI_F * ((float)m * fv);
                    float sb = __sinf(xb), cb = __cosf(xb);
                    float ss = __sinf(xs), cs = __cosf(xs);
                    g = atan2f(sb + ss, cb + cs) * INV_PI;
                }
                out_g[(size_t)m * DIM_VSA + col] = g;
            }
        }
    }
}

extern "C" void kernel_launch(void* const* d_in, const int* in_sizes, int n_in,
                              void* d_out, int out_size, void* d_ws, size_t ws_size,
                              hipStream_t stream) {
    const float*         atoms = (const float*)d_in[0];
    const float*         pairs = (const float*)d_in[1];
    const unsigned char* mask  = (const unsigned char*)d_in[2];
    // d_in[3] = `active` (unused by the reference computation)
    const float*         atomP = (const float*)d_in[4];
    const float*         bondP = (const float*)d_in[5];
    const float*         freq  = (const float*)d_in[6];
    float*               out   = (float*)d_out;

    const int batch = in_sizes[0] / (N_ATOMS * DIM_NODE);
    graph_encode_kernel<<<batch, 256, 0, stream>>>(atoms, pairs, mask, atomP, bondP, freq, out);
}